// Synapse_66400194396810
// MI455X (gfx1250) — compile-verified
//
#include <hip/hip_runtime.h>
#include <stdint.h>
#include <math.h>

// ---------------- tiling ----------------
#define NH 8
#define TPB 256                     // 8 wave32s per block
#define CPT 4                       // columns per thread (float4)
#define TILE_COLS (TPB * CPT)       // 1024 columns per block
#define TILE_ROWS 8                 // rows per TDM tile (32 KB)
#define RB_PER_BLOCK 4              // row-blocks per block -> 32 rows/block
#define ROWS_PER_BLOCK (TILE_ROWS * RB_PER_BLOCK)

typedef __attribute__((ext_vector_type(4))) unsigned int u32x4;
typedef __attribute__((ext_vector_type(8))) int          i32x8;
typedef __attribute__((ext_vector_type(4))) int          i32x4;
typedef __attribute__((ext_vector_type(2))) float        v2f;
typedef __attribute__((ext_vector_type(8))) float        v8f;

#if defined(__HIP_DEVICE_COMPILE__) && __has_builtin(__builtin_amdgcn_tensor_load_to_lds)
#define USE_TDM 1
#else
#define USE_TDM 0
#endif

#if defined(__HIP_DEVICE_COMPILE__) && __has_builtin(__builtin_amdgcn_wmma_f32_16x16x4_f32)
#define USE_WMMA 1
#else
#define USE_WMMA 0
#endif

// Native gfx1250 tanh (1 trans op). Fallback: exp2+rcp (2 trans ops).
__device__ __forceinline__ float fast_tanh(float x) {
#if defined(__HIP_DEVICE_COMPILE__) && __has_builtin(__builtin_amdgcn_tanhf)
    return __builtin_amdgcn_tanhf(x);
#elif defined(__HIP_DEVICE_COMPILE__)
    float e = __builtin_amdgcn_exp2f(x * 2.88539008177792681472f); // 2*log2(e)
    return 1.0f - 2.0f * __builtin_amdgcn_rcpf(e + 1.0f);
#else
    return tanhf(x);
#endif
}

#if USE_TDM
// Tensor Data Mover: DMA a TILE_ROWS x TILE_COLS f32 tile (row-major, row
// stride = tensor_w elements) from global memory into LDS. D# per ISA ch.8.
__device__ __forceinline__ void tdm_load_tile(const float* gsrc, unsigned lds_off,
                                              int tensor_w, int tensor_h) {
    unsigned long long ga = (unsigned long long)(uintptr_t)gsrc;
    u32x4 g0;
    g0[0] = 1u;                                             // count=1, user desc
    g0[1] = lds_off;                                        // lds_addr (bytes)
    g0[2] = (unsigned)(ga & 0xFFFFFFFFull);                 // global_addr[31:0]
    g0[3] = (unsigned)((ga >> 32) & 0x01FFFFFFull)          // global_addr[56:32]
            | 0x80000000u;                                  // type=2 ("image")
    i32x8 g1;
    g1[0] = (int)(2u << 16);                                // data_size=2 -> 4 bytes
    g1[1] = (int)(((unsigned)tensor_w & 0xFFFFu) << 16);    // tensor_dim0[15:0]
    g1[2] = (int)((((unsigned)tensor_w >> 16) & 0xFFFFu)    // tensor_dim0[31:16]
                | (((unsigned)tensor_h & 0xFFFFu) << 16));  // tensor_dim1[15:0]
    g1[3] = (int)((((unsigned)tensor_h >> 16) & 0xFFFFu)    // tensor_dim1[31:16]
                | ((unsigned)TILE_COLS << 16));             // tile_dim0
    g1[4] = TILE_ROWS;                                      // tile_dim1 (tile_dim2=0)
    g1[5] = tensor_w;                                       // tensor_dim0_stride[31:0]
    g1[6] = 0;                                              // stride hi / dim1_stride
    g1[7] = 0;
    i32x4 z4; z4[0] = 0; z4[1] = 0; z4[2] = 0; z4[3] = 0;   // groups 2/3 unused (2D)
    i32x8 z8; z8[0] = 0; z8[1] = 0; z8[2] = 0; z8[3] = 0;
              z8[4] = 0; z8[5] = 0; z8[6] = 0; z8[7] = 0;
    __builtin_amdgcn_tensor_load_to_lds(g0, g1, z4, z4, z8, 0);
}
#endif

__global__ __launch_bounds__(TPB) void synapse_mlp_kernel(
    const float* __restrict__ A, const float* __restrict__ pre,
    const float* __restrict__ post, const float* __restrict__ w1,
    const float* __restrict__ b1, const float* __restrict__ w2,
    const float* __restrict__ b2, float* __restrict__ out, int M, int N)
{
    __shared__ float tile[2][TILE_ROWS][TILE_COLS];         // 64 KB double buffer
    __shared__ float s1buf[ROWS_PER_BLOCK][16];             // 2 KB row factors

    const int tid  = threadIdx.x;
    const int col0 = blockIdx.x * TILE_COLS;
    const int row0 = blockIdx.y * ROWS_PER_BLOCK;
    const int c    = col0 + tid * CPT;

    // Tiny uniform weights -> registers (scalar loads).
    float w0[NH], wpre[NH], u[NH];
#pragma unroll
    for (int h = 0; h < NH; ++h) {
        w0[h]   = w1[h * 3 + 0];
        wpre[h] = w1[h * 3 + 1];
        u[h]    = w2[h];
    }
    const float b2s = b2[0];

    // ---- Row-factor precompute s1[m][h] = post[m]*w1[h,2] + b1[h] ----
    // Rank-1 + bias == (16x4)x(4x16) GEMM -> V_WMMA_F32_16X16X4_F32.
    // Done ONCE per block by wave 0 on the matrix pipe (2 WMMAs for 32 rows),
    // replacing 8 redundant FMAs per row in every thread with an LDS broadcast.
#if USE_WMMA
    if (tid < 32) {
        const int l = tid;                                  // lane, EXEC all-1s
        // B 4x16: lanes 0-15 -> {K0,K1} rows, lanes 16-31 -> {K2,K3}=0, N=l&15
        v2f bv; bv[0] = 0.0f; bv[1] = 0.0f;
        if (l < NH) { bv[0] = w1[3 * l + 2]; bv[1] = b1[l]; }
#pragma unroll
        for (int g = 0; g < ROWS_PER_BLOCK / 16; ++g) {
            // A 16x4: lanes 0-15 -> {K0,K1}={post[m],1}, lanes 16-31 -> {K2,K3}=0
            v2f av;
            av[0] = (l < 16) ? post[row0 + g * 16 + l] : 0.0f;
            av[1] = (l < 16) ? 1.0f : 0.0f;
            v8f d = {};
            d = __builtin_amdgcn_wmma_f32_16x16x4_f32(
                    false, av, false, bv, (short)0, d, false, false);
            // D layout: VGPR v, lane l -> M = v + 8*(l>=16), N = l&15
            const int mbase = g * 16 + ((l >= 16) ? 8 : 0);
            const int n = l & 15;
#pragma unroll
            for (int v = 0; v < 8; ++v) s1buf[mbase + v][n] = d[v];
        }
    }
#else
    // Fallback: first NH*ROWS_PER_BLOCK threads compute s1 directly.
    if (tid < ROWS_PER_BLOCK * NH) {
        const int m = tid / NH, h = tid % NH;
        s1buf[m][h] = fmaf(post[row0 + m], w1[3 * h + 2], b1[h]);
    }
#endif

    // Rank-1 column factor: c1[j][h] = pre[n]*w1[h,1], amortized over all rows.
    const float4 p4 = *(const float4*)(pre + c);
    const float pv[CPT] = {p4.x, p4.y, p4.z, p4.w};
    float c1[CPT][NH];
#pragma unroll
    for (int j = 0; j < CPT; ++j)
#pragma unroll
        for (int h = 0; h < NH; ++h) c1[j][h] = pv[j] * wpre[h];

    auto stage = [&](int rb, int buf) {
        const float* src = A + (size_t)(row0 + rb * TILE_ROWS) * N + col0;
#if USE_TDM
        if (tid < 32)  // one wave drives the TDM for the whole block
            tdm_load_tile(src, (unsigned)(uintptr_t)&tile[buf][0][0], N, M);
#else
#pragma unroll
        for (int r = 0; r < TILE_ROWS; ++r)
            *(float4*)&tile[buf][r][tid * CPT] =
                *(const float4*)(src + (size_t)r * N + tid * CPT);
#endif
    };

    stage(0, 0);
    for (int rb = 0; rb < RB_PER_BLOCK; ++rb) {
        const int buf = rb & 1;
        // Issue-before-wait keeps the DMA engine busy; TDM ops complete in
        // order so tensorcnt<=1 implies tile rb landed. Writing buf^1 is safe:
        // all waves passed the end-of-iteration barrier of rb-1.
        if (rb + 1 < RB_PER_BLOCK) {
            stage(rb + 1, buf ^ 1);
#if USE_TDM
            if (tid < 32) __builtin_amdgcn_s_wait_tensorcnt(1);
#endif
        } else {
#if USE_TDM
            if (tid < 32) __builtin_amdgcn_s_wait_tensorcnt(0);
#endif
        }
        __syncthreads();     // tile + (first iter) s1buf visible to all waves

#pragma unroll
        for (int r = 0; r < TILE_ROWS; ++r) {
            const int m = row0 + rb * TILE_ROWS + r;
            const int mloc = rb * TILE_ROWS + r;
            // Row factor broadcast from LDS (same address in all lanes).
            const float4 s1a = *(const float4*)&s1buf[mloc][0];
            const float4 s1b = *(const float4*)&s1buf[mloc][4];
            const float s1[NH] = {s1a.x, s1a.y, s1a.z, s1a.w,
                                  s1b.x, s1b.y, s1b.z, s1b.w};

            const float4 a4 = *(const float4*)&tile[buf][r][tid * CPT];
            const float av[CPT] = {a4.x, a4.y, a4.z, a4.w};
            float ov[CPT];
#pragma unroll
            for (int j = 0; j < CPT; ++j) {
                float acc = b2s;
#pragma unroll
                for (int h = 0; h < NH; ++h) {
                    float t = fmaf(av[j], w0[h], c1[j][h] + s1[h]);
                    acc = fmaf(fast_tanh(t), u[h], acc);    // v_tanh_f32 on trans pipe
                }
                ov[j] = fast_tanh(acc);
            }
            *(float4*)(out + (size_t)m * N + c) = make_float4(ov[0], ov[1], ov[2], ov[3]);
        }
        __syncthreads();                                    // done reading buf
    }
}

extern "C" void kernel_launch(void* const* d_in, const int* in_sizes, int n_in,
                              void* d_out, int out_size, void* d_ws, size_t ws_size,
                              hipStream_t stream) {
    (void)n_in; (void)out_size; (void)d_ws; (void)ws_size;
    const float* A    = (const float*)d_in[0];
    const float* pre  = (const float*)d_in[1];
    const float* post = (const float*)d_in[2];
    const float* w1   = (const float*)d_in[3];
    const float* b1   = (const float*)d_in[4];
    const float* w2   = (const float*)d_in[5];
    const float* b2   = (const float*)d_in[6];
    float* out = (float*)d_out;

    const int N = in_sizes[1];   // len(pre)  = columns
    const int M = in_sizes[2];   // len(post) = rows
    dim3 grid((unsigned)(N / TILE_COLS), (unsigned)(M / ROWS_PER_BLOCK));
    synapse_mlp_kernel<<<grid, dim3(TPB), 0, stream>>>(A, pre, post, w1, b1, w2, b2,
                                                       out, M, N);
}